// GPKANLayer_36309653520658
// MI455X (gfx1250) — compile-verified
//
#include <hip/hip_runtime.h>
#include <stdint.h>

// Problem shape (fixed by reference setup_inputs)
#define B_SZ 2048
#define O_SZ 64
#define I_SZ 64
#define M_SZ 32

#define EPS_XVAR 1e-6f
#define EPS_QVAR 1e-5f
#define EPS_VAR  1e-5f
#define MIN_SCALE 0.1f
#define EPS_EDGE 1e-6f
#define LOG2E 1.4426950408889634f

// ---- CDNA5 fast-math helpers (native v_exp_f32 / v_log_f32, both base-2) ----
static __device__ __forceinline__ float fexp2(float a) {
#if __has_builtin(__builtin_amdgcn_exp2f)
  return __builtin_amdgcn_exp2f(a);
#else
  return exp2f(a);
#endif
}
static __device__ __forceinline__ float flog2(float a) {
#if __has_builtin(__builtin_amdgcn_logf)
  return __builtin_amdgcn_logf(a);
#else
  return log2f(a);
#endif
}

// ---- CDNA5 async global->LDS copy (ASYNCcnt-tracked DMA-style copy) ----
// per-lane: copies 16 bytes from the global address in vaddr to the LDS address in vdst. GV mode (saddr is off).
static __device__ __forceinline__ void async_b128(uint32_t lds_byte_off, const float* g) {
  asm volatile("global_load_async_to_lds_b128 %0, %1, off"
               :: "v"(lds_byte_off), "v"((uint64_t)(uintptr_t)g)
               : "memory");
}
static __device__ __forceinline__ void wait_async0() {
#if __has_builtin(__builtin_amdgcn_s_wait_asynccnt)
  __builtin_amdgcn_s_wait_asynccnt(0);
#else
  asm volatile("s_wait_asynccnt 0" ::: "memory");
#endif
}

__global__ __launch_bounds__(256) void gpkan_psi_kernel(
    const float* __restrict__ x,            // (B, I)
    const float* __restrict__ z,            // (O, I, M)
    const float* __restrict__ q_mu,         // (O, I, M)
    const float* __restrict__ q_log_var,    // (O, I, M)
    const float* __restrict__ log_scale,    // (O, I)
    const float* __restrict__ log_variance, // (O, I)
    float* __restrict__ out)                // (2, B, O) flat
{
  // Per-o parameter tiles in LDS (broadcast-read by all lanes -> no bank conflicts)
  __shared__ __align__(16) float s_z  [I_SZ * M_SZ];
  __shared__ __align__(16) float s_qmu[I_SZ * M_SZ];
  __shared__ __align__(16) float s_w  [I_SZ * M_SZ]; // staged as q_log_var, transformed in place
  __shared__ float s_a2[I_SZ]; // -0.5*log2(e)/denom[i]
  __shared__ float s_lc[I_SZ]; // log2(c0[i]) with c0 = vk*sqrt(ell^2/denom)

  const int tid = threadIdx.x;
  const int o   = blockIdx.y;
  const int b   = blockIdx.x * 256 + tid;
  const int base = o * (I_SZ * M_SZ);

  // ---- stage z, q_mu, q_log_var for this o via async b128 copies (6 per thread) ----
  for (int c = tid; c < (I_SZ * M_SZ) / 4; c += 256) {
    async_b128((uint32_t)(uintptr_t)(&s_z  [c * 4]), z         + base + c * 4);
    async_b128((uint32_t)(uintptr_t)(&s_qmu[c * 4]), q_mu      + base + c * 4);
    async_b128((uint32_t)(uintptr_t)(&s_w  [c * 4]), q_log_var + base + c * 4);
  }

  const float* __restrict__ xr = x + b * I_SZ;
  __builtin_prefetch(xr, 0, 0);   // global_prefetch_b8: warm this thread's x row

  wait_async0();      // this wave's async copies complete
  __syncthreads();    // all waves' copies visible

  // ---- transform: s_w <- max(exp(q_log_var), EPS_QVAR) + q_mu^2  (in place) ----
  for (int idx = tid; idx < I_SZ * M_SZ; idx += 256) {
    float qlv = s_w[idx];
    float qm  = s_qmu[idx];
    float qv  = fmaxf(fexp2(qlv * LOG2E), EPS_QVAR);
    s_w[idx]  = fmaf(qm, qm, qv);
  }
  // ---- per-(o,i) constants folded into a single exp2 per element ----
  if (tid < I_SZ) {
    float ls  = log_scale[o * I_SZ + tid];
    float lv  = log_variance[o * I_SZ + tid];
    float ell = fmaxf(fexp2(ls * LOG2E), MIN_SCALE);
    float ell_sq = ell * ell;
    float denom  = ell_sq + EPS_XVAR;             // x_var == EPS_XVAR everywhere
    float vk  = fmaxf(fexp2(lv * LOG2E), EPS_VAR);
    float c0  = vk * __builtin_sqrtf(ell_sq / denom);
    s_a2[tid] = (-0.5f / denom) * LOG2E;
    s_lc[tid] = flog2(c0);                        // c0 > 0 always (vk >= EPS_VAR)
  }
  __syncthreads();

  // ---- main loop: 64 i x 32 m, one v_exp_f32 + ~6 VALU per element ----
  float means = 0.0f, vars = 0.0f;
  for (int i = 0; i < I_SZ; ++i) {
    const float xb = xr[i];
    const float a2 = s_a2[i];
    const float lc = s_lc[i];
    const float4* z4 = (const float4*)&s_z  [i * M_SZ];
    const float4* q4 = (const float4*)&s_qmu[i * M_SZ];
    const float4* w4 = (const float4*)&s_w  [i * M_SZ];
    float em = 0.0f, ev = 0.0f;
#pragma unroll
    for (int m4 = 0; m4 < M_SZ / 4; ++m4) {
      const float4 zz = z4[m4];
      const float4 qq = q4[m4];
      const float4 ww = w4[m4];
      {
        float d = xb - zz.x; float e = fexp2(fmaf(d * d, a2, lc));
        em = fmaf(e, qq.x, em); ev = fmaf(e * e, ww.x, ev);
      }
      {
        float d = xb - zz.y; float e = fexp2(fmaf(d * d, a2, lc));
        em = fmaf(e, qq.y, em); ev = fmaf(e * e, ww.y, ev);
      }
      {
        float d = xb - zz.z; float e = fexp2(fmaf(d * d, a2, lc));
        em = fmaf(e, qq.z, em); ev = fmaf(e * e, ww.z, ev);
      }
      {
        float d = xb - zz.w; float e = fexp2(fmaf(d * d, a2, lc));
        em = fmaf(e, qq.w, em); ev = fmaf(e * e, ww.w, ev);
      }
    }
    means += em;
    vars  += fmaxf(ev - em * em, EPS_EDGE);   // per-(b,o,i) clamp BEFORE i-sum
  }

  // outputs: (B,O) means then (B,O) vars, concatenated flat
  out[b * O_SZ + o]                 = means;
  out[B_SZ * O_SZ + b * O_SZ + o]   = vars;
}

extern "C" void kernel_launch(void* const* d_in, const int* in_sizes, int n_in,
                              void* d_out, int out_size, void* d_ws, size_t ws_size,
                              hipStream_t stream) {
  const float* x            = (const float*)d_in[0];
  const float* z            = (const float*)d_in[1];
  const float* q_mu         = (const float*)d_in[2];
  const float* q_log_var    = (const float*)d_in[3];
  const float* log_scale    = (const float*)d_in[4];
  const float* log_variance = (const float*)d_in[5];
  float* out = (float*)d_out;

  dim3 grid(B_SZ / 256, O_SZ);   // 8 x 64 = 512 blocks, 8 waves each
  gpkan_psi_kernel<<<grid, 256, 0, stream>>>(x, z, q_mu, q_log_var,
                                             log_scale, log_variance, out);
  (void)in_sizes; (void)n_in; (void)out_size; (void)d_ws; (void)ws_size;
}